// MeanAggregator_33698313404801
// MI455X (gfx1250) — compile-verified
//
#include <hip/hip_runtime.h>

typedef __attribute__((ext_vector_type(2))) float v2f;
typedef __attribute__((ext_vector_type(8))) float v8f;

#define DFEAT 128   // feature dim
#define KNEI  32    // neighbors per node

// One wave32 per output node.
//   mean_k embed[idx[node,k], :]  ==  (A x B) with A[m,k] = 1/32 (16x4 f32),
//   accumulated over 8 chained V_WMMA_F32_16X16X4_F32 per 16-feature tile.
// Lane L / L+16 always load feature (tile*16 + L) so every B column holds a
// single feature; the 4 K-rows per chunk come from 4 distinct neighbors
// (2 per lane-half), so gather traffic is exactly 32*128*4B per node.
// Features are processed in 2 passes of 4 tiles to keep accumulator VGPR
// pressure low enough for deep load pipelining.
__global__ __launch_bounds__(256) void mean_agg_wmma_kernel(
    const long long* __restrict__ neigh_idx,   // [B, 32] int64
    const float*     __restrict__ embed,       // [N, 128] f32
    float*           __restrict__ out,         // [B, 128] f32
    int nNodes)
{
  const int lane = threadIdx.x & 31;
  const int wave = threadIdx.x >> 5;
  const int node = blockIdx.x * 8 + wave;
  if (node >= nNodes) return;                  // wave-uniform guard

  // Each lane loads one neighbor id (contiguous 256B per wave).
  const long long myIdx = neigh_idx[(size_t)node * KNEI + lane];
  // Byte offset of the row; 100000*512 < 2^31 so u32 is safe.
  const unsigned myOffB = (unsigned)myIdx * (unsigned)(DFEAT * 4);

  // Prefetch this lane's neighbor row (512B = 4 x 128B lines) into caches.
  {
    const char* rp = (const char*)embed + myOffB;
    __builtin_prefetch(rp +   0, 0, 3);
    __builtin_prefetch(rp + 128, 0, 3);
    __builtin_prefetch(rp + 256, 0, 3);
    __builtin_prefetch(rp + 384, 0, 3);
  }

  const int half = lane >> 4;   // 0: lanes 0-15, 1: lanes 16-31
  const int col  = lane & 15;   // feature column within a 16-wide tile

  // K-chunk c (4 neighbors): lanes 0-15 supply neighbors 4c+0 / 4c+1
  // (B vgpr0 / vgpr1), lanes 16-31 supply neighbors 4c+2 / 4c+3.
  // BYTE offsets with the lane's feature column folded in, so each load is
  // saddr(embed) + voffset + imm.
  const char* base = (const char*)embed;
  unsigned offA[8], offB[8];
#pragma unroll
  for (int c = 0; c < 8; ++c) {
    offA[c] = __shfl(myOffB, 4 * c + 2 * half + 0, 32) + (unsigned)(col * 4);
    offB[c] = __shfl(myOffB, 4 * c + 2 * half + 1, 32) + (unsigned)(col * 4);
  }

  // A matrix: every entry 1/32  ->  D[m,n] = (1/32) * sum_k B[k,n], exact f32.
  const v2f a = {1.0f / KNEI, 1.0f / KNEI};

  // Two feature passes of 4 tiles each (tiles = 16 features).
#pragma unroll
  for (int p = 0; p < 2; ++p) {
    const int tb = p * 4;                       // tile base for this pass

    v8f acc[4] = {};
    v2f bbuf[2][4];

    // Prologue: chunk 0's B operands.
#pragma unroll
    for (int t = 0; t < 4; ++t) {
      bbuf[0][t][0] = *(const float*)(base + ((size_t)offA[0] + (tb + t) * 64));
      bbuf[0][t][1] = *(const float*)(base + ((size_t)offB[0] + (tb + t) * 64));
    }

#pragma unroll
    for (int c = 0; c < 8; ++c) {
      const int cur = c & 1;
      const int nxt = cur ^ 1;
      if (c < 7) {
#pragma unroll
        for (int t = 0; t < 4; ++t) {
          bbuf[nxt][t][0] =
              *(const float*)(base + ((size_t)offA[c + 1] + (tb + t) * 64));
          bbuf[nxt][t][1] =
              *(const float*)(base + ((size_t)offB[c + 1] + (tb + t) * 64));
        }
      }
#pragma unroll
      for (int t = 0; t < 4; ++t) {
        // 8 args: (neg_a, A, neg_b, B, c_mod, C, reuse_a, reuse_b)
        acc[t] = __builtin_amdgcn_wmma_f32_16x16x4_f32(
            /*neg_a=*/false, a, /*neg_b=*/false, bbuf[cur][t],
            /*c_mod=*/(short)0, acc[t], /*reuse_a=*/false, /*reuse_b=*/false);
      }
    }

    // All D rows are identical (uniform A); row M=0 lives in acc[t][0] on
    // lanes 0-15 with N = col. Lanes 0-15 write this pass's 4 tiles.
    if (half == 0) {
      float* op = out + (size_t)node * DFEAT + tb * 16 + col;
#pragma unroll
      for (int t = 0; t < 4; ++t) op[t * 16] = acc[t][0];
    }
  }
}

extern "C" void kernel_launch(void* const* d_in, const int* in_sizes, int n_in,
                              void* d_out, int out_size, void* d_ws, size_t ws_size,
                              hipStream_t stream) {
  (void)n_in; (void)out_size; (void)d_ws; (void)ws_size;
  const long long* neigh = (const long long*)d_in[0];  // int64 indices
  const float*     embed = (const float*)d_in[1];      // f32 table
  float*           out   = (float*)d_out;              // f32 [B,128]

  const int nNodes = in_sizes[0] / KNEI;               // 50000
  const int blocks = (nNodes + 7) / 8;                 // 8 waves (nodes) / block
  mean_agg_wmma_kernel<<<blocks, 256, 0, stream>>>(neigh, embed, out, nNodes);
}